// OPTAttention_Mask_cam_7026566496519
// MI455X (gfx1250) — compile-verified
//
#include <hip/hip_runtime.h>
#include <math.h>
#include <stdint.h>

typedef __attribute__((ext_vector_type(2))) float        v2f;
typedef __attribute__((ext_vector_type(8))) float        v8f;
typedef __attribute__((ext_vector_type(4))) unsigned int v4u;
typedef __attribute__((ext_vector_type(4))) int          v4i;
typedef __attribute__((ext_vector_type(8))) int          v8i;

#define E     4096
#define HD    128
#define NH    32
#define NB    4
#define PAST  2048
#define SEQ   2049
#define SB    204
#define RB    204
#define NACT  410        // (SB+1) + (RB+1) live columns
#define MERGE 32

#define KSPL  4          // K split across gridDim.y
#define KWIN  (E / KSPL) // 1024 K per block
#define CH    32         // K chunk per TDM tile (floats)
#define NCHK  (KWIN / CH)
#define WROWS 16
#define WSTR  (CH + 4)   // LDS row stride after TDM padding (bank-conflict free)
#define XSTR  1028       // X slice LDS row stride (bank-conflict free)

#if __has_builtin(__builtin_amdgcn_tensor_load_to_lds)
#define HAS_TDM 1
#else
#define HAS_TDM 0
#endif

#if HAS_TDM
// D# group1: data_size=4B, pad_enable, pad_interval=32 dwords (code 4),
// pad_amount=4 dwords (code 3); tensor = [4096 x 4096], tile = 32 x 16,
// tensor_dim0_stride = 4096.
__device__ __forceinline__ void tdm_issue(unsigned lds_addr, unsigned long long ga) {
    const v8i g1 = {
        (int)((2u << 16) | (1u << 20) | (4u << 22) | (3u << 25)),
        (int)(((unsigned)E & 0xFFFFu) << 16),                       // dim0 lo16
        (int)(((unsigned)E & 0xFFFFu) << 16),                       // dim0 hi|dim1 lo
        (int)((unsigned)CH << 16),                                  // tile_dim0
        (int)((unsigned)WROWS),                                     // tile_dim1
        (int)E,                                                     // dim0_stride lo
        0, 0
    };
    v4u g0;
    g0.x = 1u;                                   // count=1, user mode
    g0.y = lds_addr;                             // LDS byte address
    g0.z = (unsigned)ga;                         // global addr lo
    g0.w = (unsigned)(ga >> 32) | 0x80000000u;   // addr hi | type=2 (bits 127:126)
    const v4i z4 = {0, 0, 0, 0};
#if defined(__clang_major__) && (__clang_major__ >= 23)
    const v8i z8 = {0, 0, 0, 0, 0, 0, 0, 0};
    __builtin_amdgcn_tensor_load_to_lds(g0, g1, z4, z4, z8, 0);
#else
    __builtin_amdgcn_tensor_load_to_lds(g0, g1, z4, z4, 0);
#endif
}
#else
// Fallback: synchronous per-wave copy of one 16x32 W tile into padded LDS.
__device__ __forceinline__ void tile_copy(float* dst, const float* src) {
    const int lane = threadIdx.x & 31;
    const int r = lane & 15, seg = (lane >> 4) * 16;   // 16 floats per lane
    #pragma unroll
    for (int q = 0; q < 16; q += 4)
        *(float4*)&dst[r * WSTR + seg + q] = *(const float4*)&src[(size_t)r * E + seg + q];
}
#endif

// -----------------------------------------------------------------------------
// part[gy][4][E] = X[4,KWIN-slice] @ W^T-slice  via TDM->LDS->WMMA pipeline.
// Block: 4 waves, each wave owns 16 output columns across the block K-window.
// -----------------------------------------------------------------------------
__global__ __launch_bounds__(128) void gemm_tdm(
    const float* __restrict__ X,
    const float* __restrict__ W,
    float* __restrict__ part)
{
    __shared__ float xlds[4 * XSTR];                 // X slice (4 rows x 1024)
    __shared__ float wlds[4 * 2 * WROWS * WSTR];     // per-wave double buffers

    const int t = threadIdx.x, lane = t & 31, wave = t >> 5;
    const int kbase = blockIdx.y * KWIN;

    // stage X[0..3][kbase..kbase+KWIN) into LDS
    for (int i = t * 4; i < 4 * KWIN; i += 128 * 4) {
        const int r = i >> 10, kk = i & (KWIN - 1);
        *(float4*)&xlds[r * XSTR + kk] = *(const float4*)&X[(size_t)r * E + kbase + kk];
    }
    __syncthreads();

    const int m    = lane & 15;
    const int koff = (lane < 16) ? 0 : 2;
    const int n0   = blockIdx.x * 64 + wave * 16;    // this wave's 16 W rows
    float* wbuf[2] = { &wlds[(wave * 2 + 0) * WROWS * WSTR],
                       &wlds[(wave * 2 + 1) * WROWS * WSTR] };
#if HAS_TDM
    const unsigned long long gbase =
        (unsigned long long)(uintptr_t)(W + (size_t)n0 * E + kbase);
    const unsigned ldsa[2] = { (unsigned)(uintptr_t)wbuf[0],
                               (unsigned)(uintptr_t)wbuf[1] };
    tdm_issue(ldsa[0], gbase);                       // prologue: chunk 0
#else
    const float* gsrc = W + (size_t)n0 * E + kbase;
    tile_copy(wbuf[0], gsrc);
#endif

    v8f c0 = {}; v8f c1 = {};
    const float* xb0 = &xlds[(lane & 3) * XSTR + koff];

    for (int c = 0; c < NCHK - 1; ++c) {
#if HAS_TDM
        tdm_issue(ldsa[(c + 1) & 1], gbase + (unsigned long long)(c + 1) * (CH * 4));
        __builtin_amdgcn_s_wait_tensorcnt(1);        // oldest (chunk c) complete
#else
        tile_copy(wbuf[(c + 1) & 1], gsrc + (size_t)(c + 1) * CH);
#endif
        const float* bb = &wbuf[c & 1][m * WSTR + koff];
        const float* xb = xb0 + c * CH;
        #pragma unroll
        for (int k = 0; k < CH; k += 8) {
            const v2f a0 = *(const v2f*)(xb + k);
            const v2f a1 = *(const v2f*)(xb + k + 4);
            const v2f b0 = *(const v2f*)(bb + k);
            const v2f b1 = *(const v2f*)(bb + k + 4);
            c0 = __builtin_amdgcn_wmma_f32_16x16x4_f32(false, a0, false, b0,
                                                       (short)0, c0, false, false);
            c1 = __builtin_amdgcn_wmma_f32_16x16x4_f32(false, a1, false, b1,
                                                       (short)0, c1, false, false);
        }
    }
#if HAS_TDM
    __builtin_amdgcn_s_wait_tensorcnt(0);            // last chunk landed
#endif
    {
        const int c = NCHK - 1;
        const float* bb = &wbuf[c & 1][m * WSTR + koff];
        const float* xb = xb0 + c * CH;
        #pragma unroll
        for (int k = 0; k < CH; k += 8) {
            const v2f a0 = *(const v2f*)(xb + k);
            const v2f a1 = *(const v2f*)(xb + k + 4);
            const v2f b0 = *(const v2f*)(bb + k);
            const v2f b1 = *(const v2f*)(bb + k + 4);
            c0 = __builtin_amdgcn_wmma_f32_16x16x4_f32(false, a0, false, b0,
                                                       (short)0, c0, false, false);
            c1 = __builtin_amdgcn_wmma_f32_16x16x4_f32(false, a1, false, b1,
                                                       (short)0, c1, false, false);
        }
    }
    if (lane < 16) {                                 // C(m,n): vgpr=m, lane=n (m<4)
        #pragma unroll
        for (int i = 0; i < 4; ++i)
            part[((size_t)blockIdx.y * 4 + i) * E + n0 + lane] = c0[i] + c1[i];
    }
}

// Y[4,E] = (sum_s part[s] + bias) * scale
__global__ __launch_bounds__(256) void combine4(
    const float* __restrict__ part, const float* __restrict__ bias,
    float* __restrict__ Y, float scale)
{
    const int idx = blockIdx.x * 256 + threadIdx.x;   // over 4*E
    const int col = idx & (E - 1);
    float s = 0.0f;
    #pragma unroll
    for (int k = 0; k < KSPL; ++k) s += part[(size_t)k * 4 * E + idx];
    Y[idx] = (s + bias[col]) * scale;
}

// -----------------------------------------------------------------------------
// Per-(b,h) CAM attention: only the 410 live columns are touched.
// -----------------------------------------------------------------------------
__global__ __launch_bounds__(256) void attn_cam(
    const float* __restrict__ qws, const float* __restrict__ kws, const float* __restrict__ vws,
    const float* __restrict__ past_key, const float* __restrict__ past_value,
    const float* __restrict__ amask, const float* __restrict__ prev_scores,
    float* __restrict__ attn_out)
{
    __shared__ float sc[NACT];
    __shared__ float redsum[256];
    __shared__ float po[8 * HD];
    __shared__ float sval[4];   // [0] merge_scale, [3] msum*merge_scale

    const int bh = blockIdx.x;
    const int b  = bh >> 5;
    const int h  = bh & 31;
    const int t    = threadIdx.x;
    const int lane = t & 31;
    const int wv   = t >> 5;

    const float* ps    = prev_scores + (size_t)bh * PAST;
    const float* kpast = past_key   + (size_t)bh * PAST * HD;
    const float* vpast = past_value + (size_t)bh * PAST * HD;
    const float* qv    = qws + (size_t)b * E + h * HD;
    const float* knew  = kws + (size_t)b * E + h * HD;
    const float* vnew  = vws + (size_t)b * E + h * HD;
    const float* am    = amask + (size_t)b * SEQ;

    float ssum = 0.0f;
    for (int i = t; i < SB + RB; i += 256) {
        const int idx = (i < SB) ? i : (PAST - RB + (i - SB));
        ssum += ps[idx];
    }
    redsum[t] = ssum;
    __syncthreads();
    for (int s = 128; s > 0; s >>= 1) {
        if (t < s) redsum[t] += redsum[t + s];
        __syncthreads();
    }
    if (t == 0) {
        const float mean = redsum[0] / (float)(SB + RB);
        const float prob = ps[PAST - RB] / mean;
        sval[0] = (prob > 1.0f) ? (1.0f / (float)MERGE) : 0.0f;
    }

    const float4 q4 = *(const float4*)&qv[lane * 4];
    for (int j = wv; j < NACT; j += 8) {
        const int col = (j <= SB) ? j : (PAST - RB + (j - (SB + 1)));
        const float* kr = (col < PAST) ? &kpast[(size_t)col * HD] : knew;
        const float4 k4 = *(const float4*)&kr[lane * 4];
        float p = q4.x * k4.x + q4.y * k4.y + q4.z * k4.z + q4.w * k4.w;
        #pragma unroll
        for (int off = 16; off > 0; off >>= 1) p += __shfl_xor(p, off, 32);
        if (lane == 0) sc[j] = p + am[col];
    }
    __syncthreads();

    float mx = -3.4e38f;
    for (int j = t; j < NACT; j += 256) mx = fmaxf(mx, sc[j]);
    redsum[t] = mx;
    __syncthreads();
    for (int s = 128; s > 0; s >>= 1) {
        if (t < s) redsum[t] = fmaxf(redsum[t], redsum[t + s]);
        __syncthreads();
    }
    mx = redsum[0];
    __syncthreads();
    float es = 0.0f;
    for (int j = t; j < NACT; j += 256) {
        const float e = __expf(sc[j] - mx);
        sc[j] = e;
        es += e;
    }
    redsum[t] = es;
    __syncthreads();
    for (int s = 128; s > 0; s >>= 1) {
        if (t < s) redsum[t] += redsum[t + s];
        __syncthreads();
    }
    const float rinv = 1.0f / redsum[0];
    for (int j = t; j < NACT; j += 256) sc[j] *= rinv;
    __syncthreads();

    if (t == 0) sc[SB] = 0.99f;                      // pinned 1.0 x 0.99 v-scale
    if (wv == 0) {
        float p = sc[SB + 2 + lane];                 // cols 1845..1876
        #pragma unroll
        for (int off = 16; off > 0; off >>= 1) p += __shfl_xor(p, off, 32);
        if (lane == 0) sval[3] = p * sval[0];
    }
    __syncthreads();

    const int dl = (t & 31) * 4;
    const int g  = t >> 5;
    float4 acc = {0.0f, 0.0f, 0.0f, 0.0f};
    for (int j = g; j < NACT; j += 8) {
        const int col = (j <= SB) ? j : (PAST - RB + (j - (SB + 1)));
        const float* vr = (col < PAST) ? &vpast[(size_t)col * HD] : vnew;
        const float4 v4 = *(const float4*)&vr[dl];
        const float wj = sc[j];
        acc.x += wj * v4.x; acc.y += wj * v4.y;
        acc.z += wj * v4.z; acc.w += wj * v4.w;
    }
    *(float4*)&po[g * HD + dl] = acc;
    __syncthreads();
    if (t < HD) {
        float o = 0.0f;
        #pragma unroll
        for (int gg = 0; gg < 8; ++gg) o += po[gg * HD + t];
        o += sval[3] * vpast[(size_t)(PAST - RB) * HD + t];
        attn_out[(size_t)b * E + h * HD + t] = o;
    }
}

// -----------------------------------------------------------------------------
extern "C" void kernel_launch(void* const* d_in, const int* in_sizes, int n_in,
                              void* d_out, int out_size, void* d_ws, size_t ws_size,
                              hipStream_t stream) {
    (void)in_sizes; (void)n_in; (void)out_size; (void)ws_size;
    const float* hidden  = (const float*)d_in[0];
    const float* pkey    = (const float*)d_in[1];
    const float* pval    = (const float*)d_in[2];
    const float* amask   = (const float*)d_in[3];
    const float* pscores = (const float*)d_in[5];
    const float* Wq = (const float*)d_in[6];  const float* bq = (const float*)d_in[7];
    const float* Wk = (const float*)d_in[8];  const float* bk = (const float*)d_in[9];
    const float* Wv = (const float*)d_in[10]; const float* bv = (const float*)d_in[11];
    const float* Wo = (const float*)d_in[12]; const float* bo = (const float*)d_in[13];
    float* out = (float*)d_out;

    float* ws   = (float*)d_ws;
    float* qws  = ws;                 // [4,4096]
    float* kws  = ws + 1 * NB * E;
    float* vws  = ws + 2 * NB * E;
    float* aws  = ws + 3 * NB * E;
    float* part = ws + 4 * NB * E;    // [KSPL][4][E]

    const float scaling = 0.08838834764831845f;   // 128^-0.5
    const dim3 ggrid(E / 64, KSPL);

    gemm_tdm<<<ggrid, 128, 0, stream>>>(hidden, Wq, part);
    combine4<<<NB * E / 256, 256, 0, stream>>>(part, bq, qws, scaling);
    gemm_tdm<<<ggrid, 128, 0, stream>>>(hidden, Wk, part);
    combine4<<<NB * E / 256, 256, 0, stream>>>(part, bk, kws, 1.0f);
    gemm_tdm<<<ggrid, 128, 0, stream>>>(hidden, Wv, part);
    combine4<<<NB * E / 256, 256, 0, stream>>>(part, bv, vws, 1.0f);

    attn_cam<<<NB * NH, 256, 0, stream>>>(qws, kws, vws, pkey, pval, amask,
                                          pscores, aws);

    gemm_tdm<<<ggrid, 128, 0, stream>>>(aws, Wo, part);
    combine4<<<NB * E / 256, 256, 0, stream>>>(part, bo, out, 1.0f);
}